// LocationSensitiveSoftAttention_88596585382537
// MI455X (gfx1250) — compile-verified
//
#include <hip/hip_runtime.h>
#include <math.h>

typedef float v2f __attribute__((ext_vector_type(2)));
typedef float v8f __attribute__((ext_vector_type(8)));
typedef unsigned int u32x4 __attribute__((ext_vector_type(4)));
typedef int i32x4 __attribute__((ext_vector_type(4)));
typedef int i32x8 __attribute__((ext_vector_type(8)));

#define B_ 64
#define T_ 2048
#define H_ 1024
#define U_ 128
#define F_ 32
#define K_ 31
#define M_ 512
#define APITCH (M_ + 4)   // 516-dword LDS pitch -> lane stride = 4 mod 64 banks

// ---------------------------------------------------------------------------
// Kernel 1: pq = query @ Wq + bq   (B x U) — tiny, scalar FMA
// ---------------------------------------------------------------------------
__global__ void __launch_bounds__(256)
pq_kernel(const float* __restrict__ query, const float* __restrict__ Wq,
          const float* __restrict__ bq, float* __restrict__ pq) {
  int idx = blockIdx.x * 256 + threadIdx.x;   // B_*U_ total
  int b = idx >> 7;
  int u = idx & (U_ - 1);
  float acc = bq[u];
  const float* q = query + (size_t)b * H_;
  for (int h = 0; h < H_; ++h) acc = fmaf(q[h], Wq[(size_t)h * U_ + u], acc);
  pq[idx] = acc;
}

// ---------------------------------------------------------------------------
// Kernel 2: pm = memory @ Wm + bm  — (B*T x 512) x (512 x 128), fp32 WMMA.
// One block (8 waves) = one 16-row strip; wave w owns col-tile w.
// The 16x512 fp32 A strip is DMA'd into LDS by the Tensor Data Mover:
// wave 0 issues a single tensor_load_to_lds descriptor (iterate mode,
// 16 iterations of a 1x512 row, lds increment 516 dwords to keep the
// bank-spreading pitch), waits on TENSORcnt, then the block barriers.
// Descriptor bit packing follows CDNA5 ISA §8 (D# groups 0-3).
// ---------------------------------------------------------------------------
__global__ void __launch_bounds__(256)
pm_kernel(const float* __restrict__ memory, const float* __restrict__ Wm,
          const float* __restrict__ bm, float* __restrict__ pm) {
  __shared__ float As[16][APITCH];
  const int rowbase = blockIdx.x * 16;
  const int tid = threadIdx.x;

  if (tid < 32) {  // wave 0 issues the TDM transfer (EXEC-independent DMA)
    const unsigned long long ga =
        (unsigned long long)(const void*)(memory + (size_t)rowbase * M_);
    const unsigned int lds_off = (unsigned int)(size_t)&As[0][0];

    u32x4 g0;
    g0[0] = 1u;                                   // count=1, user descriptor
    g0[1] = lds_off;                              // lds_addr (bytes)
    g0[2] = (unsigned int)(ga & 0xFFFFFFFFu);     // global_addr[31:0]
    g0[3] = (unsigned int)((ga >> 32) & 0x01FFFFFFu)
          | 0x80000000u;                          // global_addr[56:32] | type=2

    i32x8 g1;
    g1[0] = (2 << 16) | (1 << 19);                // data_size=4B, iterate_enable
    g1[1] = (int)(512u << 16);                    // tensor_dim0[15:0]=512 @bits63:48
    g1[2] = (int)(((unsigned)(B_ * T_) & 0xFFFFu) << 16);  // tensor_dim1 lo16
    g1[3] = (int)(((unsigned)(B_ * T_) >> 16) & 0xFFFFu)   // tensor_dim1 hi16
          | (int)(512u << 16);                    // tile_dim0 = 512
    g1[4] = 1;                                    // tile_dim1=1 (row), tile_dim2=0
    g1[5] = 512;                                  // tensor_dim0_stride lo32
    g1[6] = 0;                                    // stride hi | dim1_stride lo
    g1[7] = 0;

    i32x4 g2;
    g2[0] = 0;                                    // tensor_dim2 (unused)
    g2[1] = APITCH;                               // lds_addr_increment = 516 elems
    g2[2] = 512;                                  // global_addr_increment lo32
    g2[3] = (int)(15u << 16);                     // inc hi16=0 | iterate_count=15 (16x)
    i32x4 g3 = {0, 0, 0, 0};
    i32x8 g4 = {0, 0, 0, 0, 0, 0, 0, 0};          // extra group (clang-23 6-arg form)

    __builtin_amdgcn_tensor_load_to_lds(g0, g1, g2, g3, g4, 0);
    __builtin_amdgcn_s_wait_tensorcnt(0);
  }
  __syncthreads();

  const int wave = tid >> 5, lane = tid & 31;
  const int hi = lane >> 4, l16 = lane & 15;
  const int col = wave * 16 + l16;

  const float bmv = bm[col];
  v8f c;
  for (int i = 0; i < 8; ++i) c[i] = bmv;

  for (int k0 = 0; k0 < M_; k0 += 4) {
    const int ka = k0 + 2 * hi;                 // lane-half K split
    v2f a, b;
    a.x = As[l16][ka];                          // A: M=l16, K=ka
    a.y = As[l16][ka + 1];                      //    K=ka+1
    b.x = Wm[(size_t)ka * U_ + col];            // B: K=ka,   N=col
    b.y = Wm[(size_t)(ka + 1) * U_ + col];      //    K=ka+1
    c = __builtin_amdgcn_wmma_f32_16x16x4_f32(false, a, false, b,
                                              (short)0, c, false, false);
  }

  for (int r = 0; r < 8; ++r)
    pm[(size_t)(rowbase + r + 8 * hi) * U_ + col] = c[r];
}

// ---------------------------------------------------------------------------
// Kernel 3 (fused): dilated conv -> pl -> X = pq+pl+pm -> X@We+be
//                   -> tanh * va -> reduce over u -> energy[b,t]
// Block = 128 threads (4 waves) handles 16 consecutive t of one batch.
// ---------------------------------------------------------------------------
__global__ void __launch_bounds__(128)
energy_kernel(const float* __restrict__ state, const float* __restrict__ conv_w,
              const float* __restrict__ conv_b, const float* __restrict__ Wl,
              const float* __restrict__ bl, const float* __restrict__ We,
              const float* __restrict__ be, const float* __restrict__ va,
              const float* __restrict__ pq, const float* __restrict__ pm,
              float* __restrict__ energy) {
  __shared__ float fsh[F_][16];
  __shared__ float Xs[16][U_ + 4];   // pitch 132 -> bank spread
  __shared__ float esum[16];

  const int tid = threadIdx.x;
  const int b = blockIdx.x >> 7;            // 128 tiles of 16 per batch
  const int t0 = (blockIdx.x & 127) << 4;

  if (tid < 16) esum[tid] = 0.0f;

  // dilated conv (dilation 2, pad 30): f[c][tt]
  for (int i = tid; i < F_ * 16; i += 128) {
    const int cch = i >> 4, tt = i & 15;
    float acc = conv_b[cch];
    for (int k = 0; k < K_; ++k) {
      const int ti = t0 + tt - 30 + 2 * k;
      if ((unsigned)ti < (unsigned)T_)
        acc = fmaf(conv_w[cch * K_ + k], state[(size_t)b * T_ + ti], acc);
    }
    fsh[cch][tt] = acc;
  }
  __syncthreads();

  // X[tt][u] = pq + pl + pm
  {
    const int u = tid;
    const float pqv = pq[b * U_ + u];
    const float blv = bl[u];
    for (int tt = 0; tt < 16; ++tt) {
      float pl = blv;
      for (int cch = 0; cch < F_; ++cch)
        pl = fmaf(fsh[cch][tt], Wl[cch * U_ + u], pl);
      Xs[tt][u] = pqv + pl + pm[(size_t)(b * T_ + t0 + tt) * U_ + u];
    }
  }
  __syncthreads();

  // Y = X @ We + be, then energy partials via tanh * va
  const int wave = tid >> 5, lane = tid & 31;
  const int hi = lane >> 4, l16 = lane & 15;

  for (int ct = 2 * wave; ct < 2 * wave + 2; ++ct) {
    const int col = ct * 16 + l16;
    const float bev = be[col];
    v8f c;
    for (int i = 0; i < 8; ++i) c[i] = bev;

    for (int k0 = 0; k0 < U_; k0 += 4) {
      const int ka = k0 + 2 * hi;
      v2f a, bb;
      a.x = Xs[l16][ka];
      a.y = Xs[l16][ka + 1];
      bb.x = We[ka * U_ + col];
      bb.y = We[(ka + 1) * U_ + col];
      c = __builtin_amdgcn_wmma_f32_16x16x4_f32(false, a, false, bb,
                                                (short)0, c, false, false);
    }

    const float vav = va[col];
    for (int r = 0; r < 8; ++r) {
      float t = vav * tanhf(c[r]);
      // reduce across the 16 lanes of this half (xor<=8 stays in-half)
      t += __shfl_xor(t, 1, 32);
      t += __shfl_xor(t, 2, 32);
      t += __shfl_xor(t, 4, 32);
      t += __shfl_xor(t, 8, 32);
      if (l16 == 0) atomicAdd(&esum[r + 8 * hi], t);   // ds_add_f32
    }
  }
  __syncthreads();
  if (tid < 16) energy[(size_t)b * T_ + t0 + tid] = esum[tid];
}

// ---------------------------------------------------------------------------
// Kernel 4: sg = sigmoid(energy) (in place), sums[b] = sum_t sg
// ---------------------------------------------------------------------------
__global__ void __launch_bounds__(256)
norm_kernel(float* __restrict__ energy, float* __restrict__ sums) {
  __shared__ float red[256];
  const int b = blockIdx.x, tid = threadIdx.x;
  float acc = 0.0f;
  for (int t = tid; t < T_; t += 256) {
    const float e = energy[(size_t)b * T_ + t];
    const float s = 1.0f / (1.0f + expf(-e));
    energy[(size_t)b * T_ + t] = s;   // overwrite with sg
    acc += s;
  }
  red[tid] = acc;
  __syncthreads();
  for (int s = 128; s > 0; s >>= 1) {
    if (tid < s) red[tid] += red[tid + s];
    __syncthreads();
  }
  if (tid == 0) sums[b] = red[0];
}

// ---------------------------------------------------------------------------
// Kernel 5: context[b,u] = sum_t (prev_cum + sg/sums[b]) * pm[b,t,u]
// ---------------------------------------------------------------------------
__global__ void __launch_bounds__(128)
context_kernel(const float* __restrict__ prev_cum, const float* __restrict__ sg,
               const float* __restrict__ sums, const float* __restrict__ pm,
               float* __restrict__ out) {
  const int b = blockIdx.x, u = threadIdx.x;
  const float inv = 1.0f / sums[b];
  float acc = 0.0f;
  for (int t = 0; t < T_; ++t) {
    const float w = prev_cum[(size_t)b * T_ + t] + sg[(size_t)b * T_ + t] * inv;
    acc = fmaf(w, pm[(size_t)(b * T_ + t) * U_ + u], acc);
  }
  out[b * U_ + u] = acc;
}

// ---------------------------------------------------------------------------
extern "C" void kernel_launch(void* const* d_in, const int* in_sizes, int n_in,
                              void* d_out, int out_size, void* d_ws, size_t ws_size,
                              hipStream_t stream) {
  const float* query    = (const float*)d_in[0];
  const float* state    = (const float*)d_in[1];
  const float* prev_cum = (const float*)d_in[2];
  const float* memory   = (const float*)d_in[3];
  const float* Wq       = (const float*)d_in[4];
  const float* bq       = (const float*)d_in[5];
  const float* Wm       = (const float*)d_in[6];
  const float* bm       = (const float*)d_in[7];
  const float* Wl       = (const float*)d_in[8];
  const float* bl       = (const float*)d_in[9];
  const float* We       = (const float*)d_in[10];
  const float* be       = (const float*)d_in[11];
  const float* va       = (const float*)d_in[12];
  const float* conv_w   = (const float*)d_in[13];
  const float* conv_b   = (const float*)d_in[14];

  float* ws     = (float*)d_ws;
  float* pq     = ws;                               // B*U          = 8192
  float* pm     = pq + (size_t)B_ * U_;             // B*T*U        = 16M floats
  float* energy = pm + (size_t)B_ * T_ * U_;        // B*T          = 131072
  float* sums   = energy + (size_t)B_ * T_;         // B            = 64
  float* out    = (float*)d_out;                    // B*U

  pq_kernel     <<<(B_ * U_) / 256, 256, 0, stream>>>(query, Wq, bq, pq);
  pm_kernel     <<<(B_ * T_) / 16,  256, 0, stream>>>(memory, Wm, bm, pm);
  energy_kernel <<<(B_ * T_) / 16,  128, 0, stream>>>(state, conv_w, conv_b, Wl, bl,
                                                      We, be, va, pq, pm, energy);
  norm_kernel   <<<B_, 256, 0, stream>>>(energy, sums);
  context_kernel<<<B_, 128, 0, stream>>>(prev_cum, energy, sums, pm, out);
}